// MHLMachine_61040075210843
// MI455X (gfx1250) — compile-verified
//
#include <hip/hip_runtime.h>

// fp32 WMMA pipeline for the Toeplitz-attention reference (MI455X / gfx1250).
// Bandwidth-bound problem (~0.1 TFLOP vs ~1 GB traffic @ 23.3 TB/s) -> keep
// fp32 end-to-end, use V_WMMA_F32_16X16X4_F32 for all GEMM stages, and focus
// on the data path: TDM tensor_load_to_lds for A tiles, fragment-adjacent LDS
// layout for B (single ds_load_b64 per fragment), global_prefetch for the
// next k-chunk.
//
// Workspace (floats): q 8.4M | v 8.4M | w_full 67.1M | w_attn 33.6M |
// rowsum 32K | attn 8.4M  (~482 MB total)

typedef float v2f __attribute__((ext_vector_type(2)));
typedef float v8f __attribute__((ext_vector_type(8)));
typedef unsigned int u32x4 __attribute__((ext_vector_type(4)));
typedef int i32x4 __attribute__((ext_vector_type(4)));
typedef int i32x8 __attribute__((ext_vector_type(8)));

#define TM 128
#define TN 128
#define KC 16
#define LDA_S 20            // padded LDS row stride for A (16 + 4 pad DWORDs)

#if defined(__has_builtin)
#if __has_builtin(__builtin_amdgcn_tensor_load_to_lds)
#define HAVE_TDM 1
#endif
#if __has_builtin(__builtin_amdgcn_s_wait_tensorcnt)
#define HAVE_TENSORCNT 1
#endif
#endif

// Generic batched GEMM: C[z] = A[z] (MxK) * B[z] (KxN), optional B transpose
// (B[k,n] = Bm[n*ldb+k]) and optional row-normalization epilogue
// (C[m,:] /= rowScale[m] + 1e-8). z decomposes as (b,h) with h = z % Hn.
__global__ __launch_bounds__(256) void wmma_gemm_f32(
    const float* __restrict__ A, int lda, long aOffB, long aOffH,
    const float* __restrict__ Bm, int ldb, long bOffB, long bOffH, int transB,
    float* __restrict__ C, int ldc, long cOffB, long cOffH,
    int M, int N, int K, int Hn,
    const float* __restrict__ rowScale, long sOffB, long sOffH)
{
    const int z  = blockIdx.z;
    const int bb = z / Hn, hh = z % Hn;
    A  += (long)bb * aOffB + (long)hh * aOffH;
    Bm += (long)bb * bOffB + (long)hh * bOffH;
    C  += (long)bb * cOffB + (long)hh * cOffH;

    __shared__ float sA[TM * LDA_S];          // [row][k], row stride 20
    __shared__ float sB[(KC / 4) * TN * 4];   // [k/4][n][k%4] fragment-adjacent

    const int tid   = threadIdx.x;
    const int lane  = tid & 31;
    const int wave  = tid >> 5;
    const int wm    = wave & 3;        // 4 waves along M
    const int wn    = wave >> 2;       // 2 waves along N
    const int m0    = blockIdx.y * TM;
    const int n0    = blockIdx.x * TN;
    const int laneM = lane & 15;
    const int kHi   = (lane >> 4) << 1;   // 0 for lanes 0-15, 2 for lanes 16-31

    v8f acc[2][4];
#pragma unroll
    for (int a = 0; a < 2; ++a)
#pragma unroll
        for (int b = 0; b < 4; ++b)
#pragma unroll
            for (int r = 0; r < 8; ++r) acc[a][b][r] = 0.f;

    for (int k0 = 0; k0 < K; k0 += KC) {
        // ---- stage A tile (TM x KC) into sA[r*20 + c] ----
#if defined(HAVE_TDM)
        if (wave == 0) {
            // Tensor DMA: 2D tile, dim0 = 16 elems along a row (64B), dim1 =
            // 128 rows, row stride = lda elements; LDS padding of 4 DWORDs
            // after every 16 DWORDs reproduces the 20-float LDS row stride.
            unsigned long long ga =
                (unsigned long long)(const void*)(A + (long)m0 * lda + k0);
            unsigned ldsa = (unsigned)(unsigned long long)(const void*)&sA[0];
            unsigned td0 = (unsigned)(K - k0);
            unsigned td1 = (unsigned)(M - m0);
            u32x4 g0;
            g0.x = 1u;                                   // count=1, no gather
            g0.y = ldsa;                                 // lds_addr
            g0.z = (unsigned)(ga & 0xFFFFFFFFu);         // global_addr[31:0]
            g0.w = (unsigned)((ga >> 32) & 0x01FFFFFFu)  // global_addr[56:32]
                   | (2u << 30);                         // type = 2 (image)
            i32x8 g1;
            g1[0] = (int)((2u << 16)      // data_size = 4B
                          | (1u << 20)    // pad_enable
                          | (3u << 22)    // pad_interval: 16 DWORDs
                          | (3u << 25));  // pad_amount:   4 DWORDs
            g1[1] = (int)((td0 & 0xFFFFu) << 16);                    // dim0 lo
            g1[2] = (int)(((td0 >> 16) & 0xFFFFu) |
                          ((td1 & 0xFFFFu) << 16));                  // dim0 hi | dim1 lo
            g1[3] = (int)(((td1 >> 16) & 0xFFFFu) |
                          ((unsigned)KC << 16));                     // dim1 hi | tile_dim0
            g1[4] = (int)(unsigned)TM;                               // tile_dim1 (tile_dim2=0)
            g1[5] = (int)(unsigned)lda;                              // dim0_stride[31:0]
            g1[6] = 0;                                               // stride hi bits
            g1[7] = 0;
            i32x4 gz4 = {0, 0, 0, 0};
            i32x8 gz8 = {0, 0, 0, 0, 0, 0, 0, 0};
            __builtin_amdgcn_tensor_load_to_lds(g0, g1, gz4, gz4, gz8, 0);
        }
#else
#pragma unroll
        for (int i = 0; i < 2; ++i) {
            int f4 = tid * 2 + i;
            int r  = f4 >> 2;
            int c  = (f4 & 3) << 2;
            int gm = m0 + r, gk = k0 + c;
            float4 val = make_float4(0.f, 0.f, 0.f, 0.f);
            if (gm < M && gk + 3 < K)
                val = *reinterpret_cast<const float4*>(A + (long)gm * lda + gk);
            sA[r * LDA_S + c + 0] = val.x;
            sA[r * LDA_S + c + 1] = val.y;
            sA[r * LDA_S + c + 2] = val.z;
            sA[r * LDA_S + c + 3] = val.w;
        }
#endif
        // ---- stage B tile (KC x TN) into sB[(k>>2)*512 + n*4 + (k&3)] ----
        if (transB) {
#pragma unroll
            for (int it = 0; it < 8; ++it) {
                int idx = tid + (it << 8);
                int k = idx & 15, n = idx >> 4;     // coalesced along k
                int gn = n0 + n, gk = k0 + k;
                float val = 0.f;
                if (gn < N && gk < K) val = Bm[(long)gn * ldb + gk];
                sB[((k >> 2) * TN + n) * 4 + (k & 3)] = val;
            }
        } else {
#pragma unroll
            for (int it = 0; it < 8; ++it) {
                int idx = tid + (it << 8);
                int n = idx & 127, k = idx >> 7;    // coalesced along n
                int gn = n0 + n, gk = k0 + k;
                float val = 0.f;
                if (gn < N && gk < K) val = Bm[(long)gk * ldb + gn];
                sB[((k >> 2) * TN + n) * 4 + (k & 3)] = val;
            }
        }

        // ---- prefetch next k-chunk while this one is consumed ----
        if (k0 + KC < K) {
            __builtin_prefetch(A + (long)(m0 + (tid >> 1)) * lda + k0 + KC, 0, 0);
            if (transB)
                __builtin_prefetch(Bm + (long)(n0 + (tid & 127)) * ldb + k0 + KC, 0, 0);
            else
                __builtin_prefetch(Bm + (long)(k0 + KC + (tid >> 4)) * ldb + n0 +
                                       (tid & 15) * 8, 0, 0);
        }

#if defined(HAVE_TDM)
        if (wave == 0) {
#if defined(HAVE_TENSORCNT)
            __builtin_amdgcn_s_wait_tensorcnt((short)0);
#else
            asm volatile("s_wait_tensorcnt 0x0" ::: "memory");
#endif
        }
#endif
        __syncthreads();

        // ---- 4 WMMA K-steps per chunk ----
#pragma unroll
        for (int kk = 0; kk < KC; kk += 4) {
            v2f afr[2];
#pragma unroll
            for (int mt = 0; mt < 2; ++mt) {
                int m = wm * 32 + mt * 16 + laneM;
                afr[mt].x = sA[m * LDA_S + kk + kHi];
                afr[mt].y = sA[m * LDA_S + kk + kHi + 1];
            }
#pragma unroll
            for (int nt = 0; nt < 4; ++nt) {
                int n = wn * 64 + nt * 16 + laneM;
                int bbase = ((kk >> 2) * TN + n) * 4 + kHi;
                v2f bfr;
                bfr.x = sB[bbase];
                bfr.y = sB[bbase + 1];
#pragma unroll
                for (int mt = 0; mt < 2; ++mt)
                    acc[mt][nt] = __builtin_amdgcn_wmma_f32_16x16x4_f32(
                        false, afr[mt], false, bfr, (short)0, acc[mt][nt],
                        false, false);
            }
        }
        __syncthreads();
    }

    const float* rs =
        rowScale ? (rowScale + (long)bb * sOffB + (long)hh * sOffH) : nullptr;
#pragma unroll
    for (int mt = 0; mt < 2; ++mt)
#pragma unroll
        for (int nt = 0; nt < 4; ++nt)
#pragma unroll
            for (int r = 0; r < 8; ++r) {
                int m = m0 + wm * 32 + mt * 16 + ((lane < 16) ? r : r + 8);
                int n = n0 + wn * 64 + nt * 16 + laneM;
                if (m < M && n < N) {
                    float val = acc[mt][nt][r];
                    if (rs) val /= (rs[m] + 1e-8f);
                    C[(long)m * ldc + n] = val;
                }
            }
}

// Toeplitz gather + cumsum over l + relu + causal mask.
// One thread per output column j; loads along a diagonal of w_full are
// coalesced in i for fixed l.
__global__ __launch_bounds__(256) void toeplitz_cumsum_kernel(
    const float* __restrict__ wfull, float* __restrict__ wattn,
    int L, int ldw)
{
    const int z = blockIdx.y;  // b*H + h
    const int j = blockIdx.x * blockDim.x + threadIdx.x;
    const float* wf = wfull + (long)z * L * ldw;
    float* wa       = wattn + (long)z * L * L;
    float acc = 0.f;
    for (int l = 0; l < L; ++l) {
        int i = (L - 1) - l + j;           // always in [0, 2L-2]
        acc += wf[(long)l * ldw + i];
        float val = acc > 0.f ? acc : 0.f; // relu(cumsum)
        wa[(long)l * L + j] = (j <= l) ? val : 0.f;
    }
}

// Row sums of w_attn (one block per row).
__global__ __launch_bounds__(256) void rowsum_kernel(
    const float* __restrict__ wattn, float* __restrict__ rowsum, int L)
{
    const long row = blockIdx.x;
    const float* p = wattn + row * (long)L;
    float s = 0.f;
    for (int j = threadIdx.x; j < L; j += 256) s += p[j];
    __shared__ float red[256];
    red[threadIdx.x] = s;
    __syncthreads();
    for (int off = 128; off > 0; off >>= 1) {
        if (threadIdx.x < off) red[threadIdx.x] += red[threadIdx.x + off];
        __syncthreads();
    }
    if (threadIdx.x == 0) rowsum[row] = red[0];
}

extern "C" void kernel_launch(void* const* d_in, const int* in_sizes, int n_in,
                              void* d_out, int out_size, void* d_ws, size_t ws_size,
                              hipStream_t stream) {
    (void)in_sizes; (void)n_in; (void)out_size; (void)ws_size;
    const float* x   = (const float*)d_in[0];
    const float* gen = (const float*)d_in[1];
    const float* Wq  = (const float*)d_in[2];
    const float* Wv  = (const float*)d_in[3];
    const float* Wp  = (const float*)d_in[4];
    float* out = (float*)d_out;

    const int B = 8, L = 1024, D = 1024, H = 4, dh = 256;
    const int NI = 2 * L - 1;   // 2047
    const int LDW = 2048;       // padded stride for w_full

    float* q      = (float*)d_ws;
    float* v      = q      + (size_t)B * L * D;
    float* wfull  = v      + (size_t)B * L * D;
    float* wattn  = wfull  + (size_t)B * H * L * LDW;
    float* rowsum = wattn  + (size_t)B * H * L * L;
    float* attn   = rowsum + (size_t)B * H * L;

    dim3 blk(256);

    // 1) q = x @ Wq^T   (M=8192, N=1024, K=1024)
    wmma_gemm_f32<<<dim3(D / TN, (B * L) / TM, 1), blk, 0, stream>>>(
        x, D, 0, 0, Wq, D, 0, 0, /*transB=*/1,
        q, D, 0, 0, B * L, D, D, 1, nullptr, 0, 0);
    //    v = x @ Wv^T
    wmma_gemm_f32<<<dim3(D / TN, (B * L) / TM, 1), blk, 0, stream>>>(
        x, D, 0, 0, Wv, D, 0, 0, /*transB=*/1,
        v, D, 0, 0, B * L, D, D, 1, nullptr, 0, 0);

    // 2) w_full[b,h] = q[b,:,h,:] (L x dh) @ gen[h] (dh x 2047)
    wmma_gemm_f32<<<dim3((NI + TN - 1) / TN, L / TM, B * H), blk, 0, stream>>>(
        q, D, (long)L * D, (long)dh,
        gen, NI, 0, (long)dh * NI, /*transB=*/0,
        wfull, LDW, (long)H * L * LDW, (long)L * LDW,
        L, NI, dh, H, nullptr, 0, 0);

    // 3) Toeplitz gather + cumsum + relu + causal mask
    toeplitz_cumsum_kernel<<<dim3(L / 256, B * H), blk, 0, stream>>>(
        wfull, wattn, L, LDW);

    // 4) row sums for normalization
    rowsum_kernel<<<dim3(B * H * L), blk, 0, stream>>>(wattn, rowsum, L);

    // 5) attn[b,:,h,:] = (w_attn[b,h] @ v[b,:,h,:]) / (rowsum + 1e-8)
    wmma_gemm_f32<<<dim3((dh + TN - 1) / TN, L / TM, B * H), blk, 0, stream>>>(
        wattn, L, (long)H * L * L, (long)L * L,
        v, D, (long)L * D, (long)dh, /*transB=*/0,
        attn, D, (long)L * D, (long)dh,
        L, dh, L, H, rowsum, (long)H * L, (long)L);

    // 6) out = attn @ Wproj^T
    wmma_gemm_f32<<<dim3(D / TN, (B * L) / TM, 1), blk, 0, stream>>>(
        attn, D, 0, 0, Wp, D, 0, 0, /*transB=*/1,
        out, D, 0, 0, B * L, D, D, 1, nullptr, 0, 0);
}